// YOLOv11Loss_11836929867787
// MI455X (gfx1250) — compile-verified
//
#include <hip/hip_runtime.h>
#include <hip/hip_bf16.h>

// ---- problem constants (match reference) ----
#define NB    32            // batch
#define NA    8400          // anchors
#define NM    40            // max gt per image
#define NCL   80            // classes
#define RM    16            // reg_max
#define BA    (NB*NA)       // 268800
#define BM    (NB*NM)       // 1280
#define EPSA  1e-9f
#define CEPS  1e-7f
#define NB7   2048          // blocks for bce kernel

typedef __attribute__((ext_vector_type(2))) float v2f;
typedef __attribute__((ext_vector_type(8))) float v8f;

#if __has_builtin(__builtin_amdgcn_wmma_f32_16x16x4_f32)
#define USE_WMMA_RED 1
#else
#define USE_WMMA_RED 0
#endif

__device__ __forceinline__ float ciou_f(float b1x1,float b1y1,float b1x2,float b1y2,
                                        float b2x1,float b2y1,float b2x2,float b2y2){
  float w1=b1x2-b1x1, h1=b1y2-b1y1+CEPS;
  float w2=b2x2-b2x1, h2=b2y2-b2y1+CEPS;
  float iw=fminf(b1x2,b2x2)-fmaxf(b1x1,b2x1);
  float ih=fminf(b1y2,b2y2)-fmaxf(b1y1,b2y1);
  float inter=fmaxf(iw,0.f)*fmaxf(ih,0.f);
  float uni=w1*h1+w2*h2-inter+CEPS;
  float iou=inter/uni;
  float cw=fmaxf(b1x2,b2x2)-fminf(b1x1,b2x1);
  float ch=fmaxf(b1y2,b2y2)-fminf(b1y1,b2y1);
  float c2=cw*cw+ch*ch+CEPS;
  float dx=b2x1+b2x2-b1x1-b1x2, dy=b2y1+b2y2-b1y1-b1y2;
  float rho2=(dx*dx+dy*dy)*0.25f;
  float at=atanf(w2/h2)-atanf(w1/h1);
  float v=0.40528473456935108577f*at*at;      // 4/pi^2
  float alpha=v/(v-iou+(1.f+CEPS));
  return iou-(rho2/c2+v*alpha);
}

// align/overlap for one (b,m,a), with the reference's mgt masking applied.
__device__ __forceinline__ void pair_vals(int b,int m,int a,
    const float* __restrict__ ps,const float* __restrict__ pb,
    const float* __restrict__ anc,const float* __restrict__ strd,
    const float* __restrict__ gtb,const float* __restrict__ mgt,
    const int* __restrict__ glab,float& alv,float& ovv){
  alv=0.f; ovv=0.f;
  int bm=b*NM+m;
  if(mgt[bm]<=0.f) return;
  float g0=gtb[(size_t)bm*4+0], g1=gtb[(size_t)bm*4+1];
  float g2=gtb[(size_t)bm*4+2], g3=gtb[(size_t)bm*4+3];
  float s=strd[a];
  float px=anc[a*2+0]*s, py=anc[a*2+1]*s;
  float dmin=fminf(fminf(px-g0,py-g1),fminf(g2-px,g3-py));
  if(dmin<=EPSA) return;
  size_t pi=(size_t)b*NA+a;
  float x=ps[pi*NCL+glab[bm]];
  float sc=1.f/(1.f+expf(-x));
  float p0=pb[pi*4+0]*s, p1=pb[pi*4+1]*s, p2=pb[pi*4+2]*s, p3=pb[pi*4+3]*s;
  float ci=ciou_f(g0,g1,g2,g3,p0,p1,p2,p3);
  ovv=fmaxf(ci,0.f);
  float o2=ovv*ovv;
  alv=sqrtf(sc)*o2*o2*o2;                      // sc^0.5 * ov^6
}

__device__ __forceinline__ float blk_sum(float v, float* sh){
  int t=threadIdx.x; sh[t]=v; __syncthreads();
  for(int o=128;o>0;o>>=1){ if(t<o) sh[t]+=sh[t+o]; __syncthreads(); }
  float r=sh[0]; __syncthreads(); return r;
}

// K1: DFL decode -> pred boxes (grid units); zero mask bitmasks + pos maxima.
__global__ void k_decode(const float* __restrict__ distri, const float* __restrict__ anc,
                         float* __restrict__ pb, unsigned long long* __restrict__ mbits,
                         unsigned int* __restrict__ posAb, unsigned int* __restrict__ posOb){
  int idx=blockIdx.x*blockDim.x+threadIdx.x;
  if(idx>=BA) return;
  mbits[idx]=0ull;
  if(idx<BM){ posAb[idx]=0u; posOb[idx]=0u; }
  const float* row=distri+(size_t)idx*64;
  int a=idx%NA;
  float ax=anc[a*2+0], ay=anc[a*2+1];
  float d[4];
#pragma unroll
  for(int s=0;s<4;++s){
    float v[16];
#pragma unroll
    for(int q=0;q<4;++q){
      float4 f=*(const float4*)(row+s*16+q*4);
      v[q*4+0]=f.x; v[q*4+1]=f.y; v[q*4+2]=f.z; v[q*4+3]=f.w;
    }
    float mx=v[0];
#pragma unroll
    for(int k=1;k<16;++k) mx=fmaxf(mx,v[k]);
    float se=0.f, sw=0.f;
#pragma unroll
    for(int k=0;k<16;++k){ float e=expf(v[k]-mx); se+=e; sw+=e*(float)k; }
    d[s]=sw/se;
  }
  pb[(size_t)idx*4+0]=ax-d[0];
  pb[(size_t)idx*4+1]=ay-d[1];
  pb[(size_t)idx*4+2]=ax+d[2];
  pb[(size_t)idx*4+3]=ay+d[3];
}

// K2: fused align + top-10 per (b,m) row; set bit m in per-anchor bitmask
// (incl. the reference's non-kept -> index-0 quirk). align never materialized.
__global__ void k_topk(const float* __restrict__ ps, const float* __restrict__ pb,
                       const float* __restrict__ anc, const float* __restrict__ strd,
                       const float* __restrict__ gtb, const float* __restrict__ mgt,
                       const int* __restrict__ glab,
                       unsigned long long* __restrict__ mbits){
  __shared__ float sv[256];
  __shared__ int   si[256];
  int bm=blockIdx.x; int b=bm/NM; int m=bm%NM;
  int tid=threadIdx.x;
  float g0=gtb[(size_t)bm*4+0], g1=gtb[(size_t)bm*4+1];
  float g2=gtb[(size_t)bm*4+2], g3=gtb[(size_t)bm*4+3];
  float msk=mgt[bm]; int lab=glab[bm];
  float lv[10]; int li_[10];
#pragma unroll
  for(int k=0;k<10;++k){ lv[k]=-1.f; li_[k]=0x7fffffff; }
  for(int a=tid;a<NA;a+=256){
    float alv=0.f;
    if(msk>0.f){
      float s=strd[a];
      float px=anc[a*2+0]*s, py=anc[a*2+1]*s;
      float dmin=fminf(fminf(px-g0,py-g1),fminf(g2-px,g3-py));
      if(dmin>EPSA){
        size_t pi=(size_t)b*NA+a;
        float x=ps[pi*NCL+lab];
        float sc=1.f/(1.f+expf(-x));
        float p0=pb[pi*4+0]*s, p1=pb[pi*4+1]*s, p2=pb[pi*4+2]*s, p3=pb[pi*4+3]*s;
        float ci=ciou_f(g0,g1,g2,g3,p0,p1,p2,p3);
        float ovv=fmaxf(ci,0.f);
        float o2=ovv*ovv;
        alv=sqrtf(sc)*o2*o2*o2;
      }
    }
    if(alv>lv[9]){
      int p=9;
      while(p>0 && alv>lv[p-1]){ lv[p]=lv[p-1]; li_[p]=li_[p-1]; --p; }
      lv[p]=alv; li_[p]=a;
    }
  }
  int head=0, notKept=0;
  for(int r=0;r<10;++r){
    float v = (head<10)?lv[head]:-1.f;
    int   i = (head<10)?li_[head]:0x7fffffff;
    sv[tid]=v; si[tid]=i; __syncthreads();
    for(int o=128;o>0;o>>=1){
      if(tid<o){
        float v2=sv[tid+o]; int i2=si[tid+o];
        if(v2>sv[tid] || (v2==sv[tid] && i2<si[tid])){ sv[tid]=v2; si[tid]=i2; }
      }
      __syncthreads();
    }
    float bv=sv[0]; int bi=si[0];
    __syncthreads();
    if(head<10 && v==bv && i==bi) head++;
    if(bv>EPSA){
      if(tid==0) atomicOr(&mbits[(size_t)b*NA+bi], 1ull<<m);
    } else {
      notKept++;
    }
  }
  if(tid==0 && notKept>0){
    float s=strd[0];
    float px=anc[0]*s, py=anc[1]*s;
    float dmin=fminf(fminf(px-g0,py-g1),fminf(g2-px,g3-py));
    if(dmin>EPSA && msk>0.f) atomicOr(&mbits[(size_t)b*NA+0], 1ull<<m);
  }
}

// K3: resolve multi-assigned anchors (argmax overlap, recomputed on demand),
// emit targets + per-anchor align value; atomic-max pos_align/pos_ov (>=0 so
// uint bit-pattern max == float max, order-independent => deterministic).
__global__ void k_resolve(const float* __restrict__ ps, const float* __restrict__ pb,
                          const float* __restrict__ anc, const float* __restrict__ strd,
                          const float* __restrict__ gtb, const float* __restrict__ mgt,
                          const int* __restrict__ glab,
                          unsigned long long* __restrict__ mbits,
                          int* __restrict__ tlab, float* __restrict__ fgf,
                          float* __restrict__ tbox, float* __restrict__ aval,
                          unsigned int* __restrict__ posAb, unsigned int* __restrict__ posOb){
  int idx=blockIdx.x*blockDim.x+threadIdx.x;
  if(idx>=BA) return;
  int b=idx/NA, a=idx%NA;
  unsigned long long bits=mbits[idx];
  if(__popcll(bits)>1){
    float best=-1.f; int bmax=0;
    for(int m=0;m<NM;++m){
      float al,ov; pair_vals(b,m,a,ps,pb,anc,strd,gtb,mgt,glab,al,ov);
      if(ov>best){ best=ov; bmax=m; }      // strict > == first max (argmax)
    }
    bits=1ull<<bmax;
    mbits[idx]=bits;
  }
  int tgt=0; float av=0.f, fgv=0.f;
  if(bits){
    tgt=__ffsll((long long)bits)-1;
    float al,ov; pair_vals(b,tgt,a,ps,pb,anc,strd,gtb,mgt,glab,al,ov);
    av=al; fgv=1.f;
    int bm=b*NM+tgt;
    atomicMax(&posAb[bm], __float_as_uint(al));
    atomicMax(&posOb[bm], __float_as_uint(ov));
  }
  aval[idx]=av;
  int lab=glab[b*NM+tgt]; if(lab<0) lab=0;
  tlab[idx]=lab;
  float inv_s=1.f/strd[a];
#pragma unroll
  for(int k=0;k<4;++k) tbox[(size_t)idx*4+k]=gtb[((size_t)b*NM+tgt)*4+k]*inv_s;
  fgf[idx]=fgv;
}

// K4: per-anchor norm + tss partials (final mask_pos is one-hot per anchor).
__global__ void k_norm(const unsigned long long* __restrict__ mbits,
                       const float* __restrict__ aval,
                       const unsigned int* __restrict__ posAb,
                       const unsigned int* __restrict__ posOb,
                       const float* __restrict__ fgf,
                       float* __restrict__ normv, float* __restrict__ tssp){
  __shared__ float sh[256];
  int idx=blockIdx.x*blockDim.x+threadIdx.x;
  int b=idx/NA;
  unsigned long long bits=mbits[idx];
  float nrm=0.f;
  if(bits){
    int m=__ffsll((long long)bits)-1;
    int bm=b*NM+m;
    nrm=aval[idx]*__uint_as_float(posOb[bm])/(__uint_as_float(posAb[bm])+EPSA);
  }
  normv[idx]=nrm;
  float p=blk_sum(nrm*fgf[idx],sh);
  if(threadIdx.x==0) tssp[blockIdx.x]=p;
}

// K5: BCE over B*A*NC, float2-vectorized loads; sum accumulated on the matrix
// pipe via chained V_WMMA_F32_16X16X4_F32 with all-ones B (total = sum(C)/16).
__global__ void k_bce(const float* __restrict__ ps, const int* __restrict__ tlab,
                      const float* __restrict__ normv, const float* __restrict__ fgf,
                      float* __restrict__ bcep){
  __shared__ float sh[256];
  const long total=(long)BA*NCL;            // even
  const long step=(long)gridDim.x*blockDim.x*2;
#if USE_WMMA_RED
  v8f c={0.f,0.f,0.f,0.f,0.f,0.f,0.f,0.f};
  const v2f ones={1.f,1.f};
#else
  float acc=0.f;
#endif
  for(long s=(long)blockIdx.x*blockDim.x*2; s<total; s+=step){
    long e0=s+(long)threadIdx.x*2;
    float f0=0.f, f1=0.f;
    if(e0<total){
      float2 xv=*(const float2*)(ps+e0);
      int  c0=(int)(e0%NCL); long an0=e0/NCL;
      int  c1=c0+1; long an1=an0;
      if(c1==NCL){ c1=0; an1=an0+1; }
      float t0=(c0==tlab[an0]) ? normv[an0]*fgf[an0] : 0.f;
      float t1=(c1==tlab[an1]) ? normv[an1]*fgf[an1] : 0.f;
      f0=fmaxf(xv.x,0.f)-xv.x*t0+log1pf(expf(-fabsf(xv.x)));
      f1=fmaxf(xv.y,0.f)-xv.y*t1+log1pf(expf(-fabsf(xv.y)));
    }
#if USE_WMMA_RED
    v2f av={f0,f1};
    c=__builtin_amdgcn_wmma_f32_16x16x4_f32(false,av,false,ones,(short)0,c,false,false);
#else
    acc+=f0+f1;
#endif
  }
#if USE_WMMA_RED
  float lane=c[0]+c[1]+c[2]+c[3]+c[4]+c[5]+c[6]+c[7];
  float p=blk_sum(lane,sh)*(1.f/16.f);
#else
  float p=blk_sum(acc,sh);
#endif
  if(threadIdx.x==0) bcep[blockIdx.x]=p;
}

// K6: IoU + DFL losses over fg anchors.
__global__ void k_boxdfl(const float* __restrict__ distri, const float* __restrict__ anc,
                         const float* __restrict__ pb, const float* __restrict__ tbox,
                         const float* __restrict__ normv, const float* __restrict__ fgf,
                         float* __restrict__ ioup, float* __restrict__ dflp){
  __shared__ float sh[256];
  int idx=blockIdx.x*blockDim.x+threadIdx.x;
  float li=0.f, ld=0.f;
  if(fgf[idx]>0.f){
    float w=normv[idx];
    float p0=pb[(size_t)idx*4+0], p1=pb[(size_t)idx*4+1];
    float p2=pb[(size_t)idx*4+2], p3=pb[(size_t)idx*4+3];
    float t0=tbox[(size_t)idx*4+0], t1=tbox[(size_t)idx*4+1];
    float t2=tbox[(size_t)idx*4+2], t3=tbox[(size_t)idx*4+3];
    float iou=ciou_f(p0,p1,p2,p3,t0,t1,t2,t3);
    li=(1.f-iou)*w;
    int a=idx%NA;
    float ax=anc[a*2+0], ay=anc[a*2+1];
    float tgt[4];
    tgt[0]=fminf(fmaxf(ax-t0,0.f),14.99f);
    tgt[1]=fminf(fmaxf(ay-t1,0.f),14.99f);
    tgt[2]=fminf(fmaxf(t2-ax,0.f),14.99f);
    tgt[3]=fminf(fmaxf(t3-ay,0.f),14.99f);
    const float* row=distri+(size_t)idx*64;
    float dsum=0.f;
#pragma unroll
    for(int s=0;s<4;++s){
      float v[16];
#pragma unroll
      for(int q=0;q<4;++q){
        float4 f=*(const float4*)(row+s*16+q*4);
        v[q*4+0]=f.x; v[q*4+1]=f.y; v[q*4+2]=f.z; v[q*4+3]=f.w;
      }
      float mx=v[0];
#pragma unroll
      for(int k=1;k<16;++k) mx=fmaxf(mx,v[k]);
      float se=0.f;
#pragma unroll
      for(int k=0;k<16;++k) se+=expf(v[k]-mx);
      float lse=mx+logf(se);
      int tl=(int)tgt[s];
      float wl=(float)(tl+1)-tgt[s];
      float wr=1.f-wl;
      float lpL=v[tl]-lse, lpR=v[tl+1]-lse;
      dsum+=-(lpL*wl+lpR*wr);
    }
    ld=(dsum*0.25f)*w;
  }
  float pi_=blk_sum(li,sh);
  float pd_=blk_sum(ld,sh);
  if(threadIdx.x==0){ ioup[blockIdx.x]=pi_; dflp[blockIdx.x]=pd_; }
}

// K7: final reduction + scaling.
__global__ void k_final(const float* __restrict__ bcep, int nb,
                        const float* __restrict__ tssp, int nt,
                        const float* __restrict__ ioup, const float* __restrict__ dflp, int ni,
                        float* __restrict__ out){
  __shared__ float sh[256];
  int tid=threadIdx.x;
  float sb=0.f; for(int i=tid;i<nb;i+=256) sb+=bcep[i];
  float st=0.f; for(int i=tid;i<nt;i+=256) st+=tssp[i];
  float si_=0.f; for(int i=tid;i<ni;i+=256) si_+=ioup[i];
  float sd=0.f; for(int i=tid;i<ni;i+=256) sd+=dflp[i];
  sb=blk_sum(sb,sh);
  st=blk_sum(st,sh);
  si_=blk_sum(si_,sh);
  sd=blk_sum(sd,sh);
  if(tid==0){
    float tss=fmaxf(st,1.f);
    out[0]=si_/tss*7.5f;   // loss_iou * HYP_BOX
    out[1]=sb /tss*0.5f;   // loss_cls * HYP_CLS
    out[2]=sd /tss*1.5f;   // loss_dfl * HYP_DFL
  }
}

extern "C" void kernel_launch(void* const* d_in, const int* in_sizes, int n_in,
                              void* d_out, int out_size, void* d_ws, size_t ws_size,
                              hipStream_t stream){
  const float* ps   =(const float*)d_in[0];
  const float* pd   =(const float*)d_in[1];
  const float* anc  =(const float*)d_in[2];
  const float* strd =(const float*)d_in[3];
  const float* gtb  =(const float*)d_in[4];
  const float* mgt  =(const float*)d_in[5];
  const int*   glab =(const int*)d_in[6];
  float* out=(float*)d_out;

  char* w=(char*)d_ws;
  size_t off=0;
  auto alloc=[&](size_t bytes)->void*{
    void* p=w+off; off=(off+bytes+255)&~(size_t)255; return p;
  };
  float* pb    =(float*)alloc((size_t)BA*4*sizeof(float));
  unsigned long long* mbits=(unsigned long long*)alloc((size_t)BA*8);
  int*   tlab  =(int*)  alloc((size_t)BA*sizeof(int));
  float* fgf   =(float*)alloc((size_t)BA*sizeof(float));
  float* tbox  =(float*)alloc((size_t)BA*4*sizeof(float));
  float* normv =(float*)alloc((size_t)BA*sizeof(float));
  float* aval  =(float*)alloc((size_t)BA*sizeof(float));
  unsigned int* posAb=(unsigned int*)alloc((size_t)BM*sizeof(unsigned int));
  unsigned int* posOb=(unsigned int*)alloc((size_t)BM*sizeof(unsigned int));
  float* bcep  =(float*)alloc((size_t)NB7*sizeof(float));
  float* tssp  =(float*)alloc((size_t)(BA/256)*sizeof(float));
  float* ioup  =(float*)alloc((size_t)(BA/256)*sizeof(float));
  float* dflp  =(float*)alloc((size_t)(BA/256)*sizeof(float));

  const int TPB=256;
  const int gBA=BA/TPB;   // 1050 (exact)

  k_decode <<<gBA, TPB, 0, stream>>>(pd, anc, pb, mbits, posAb, posOb);
  k_topk   <<<BM,  TPB, 0, stream>>>(ps, pb, anc, strd, gtb, mgt, glab, mbits);
  k_resolve<<<gBA, TPB, 0, stream>>>(ps, pb, anc, strd, gtb, mgt, glab, mbits,
                                     tlab, fgf, tbox, aval, posAb, posOb);
  k_norm   <<<gBA, TPB, 0, stream>>>(mbits, aval, posAb, posOb, fgf, normv, tssp);
  k_bce    <<<NB7, TPB, 0, stream>>>(ps, tlab, normv, fgf, bcep);
  k_boxdfl <<<gBA, TPB, 0, stream>>>(pd, anc, pb, tbox, normv, fgf, ioup, dflp);
  k_final  <<<1,   TPB, 0, stream>>>(bcep, NB7, tssp, gBA, ioup, dflp, gBA, out);
}